// GATLSTM_60825326846418
// MI455X (gfx1250) — compile-verified
//
#include <hip/hip_runtime.h>
#include <math.h>

// ---------------- problem constants (match reference) ----------------
#define N_NODES 50000
#define N_EDGES 500000
#define F_IN    388
#define H_HEADS 8
#define D1      64
#define D2      128
#define EDIM    16
#define HID     32
#define HD1     (H_HEADS * D1)   // 512
#define HD2     (H_HEADS * D2)   // 1024
#define EN      (N_EDGES + N_NODES)  // edges + self loops

typedef __attribute__((ext_vector_type(2))) float v2f;
typedef __attribute__((ext_vector_type(8))) float v8f;

__device__ __forceinline__ float sigmoidf_(float x) { return 1.0f / (1.0f + __expf(-x)); }

__device__ __forceinline__ void atomicMaxF(float* addr, float val) {
  unsigned int* ua = (unsigned int*)addr;
  unsigned int old = *ua;
  while (__uint_as_float(old) < val) {
    unsigned int assumed = old;
    old = atomicCAS(ua, assumed, __float_as_uint(val));
    if (old == assumed) break;
  }
}

// ---------------- small utility kernels ----------------
__global__ void fill_kernel(float* __restrict__ p, float v, long long n) {
  long long i = (long long)blockIdx.x * blockDim.x + threadIdx.x;
  if (i < n) p[i] = v;
}

// per (edge, attr-dim): degree count + sum of edge_attr per dst
__global__ void deg_loop_kernel(const int* __restrict__ dst,
                                const float* __restrict__ edge_attr,
                                float* __restrict__ deg, float* __restrict__ lsum) {
  long long idx = (long long)blockIdx.x * blockDim.x + threadIdx.x;
  if (idx >= (long long)N_EDGES * EDIM) return;
  long long e = idx >> 4;
  int c = (int)(idx & 15);
  int t = dst[e];
  atomicAdd(&lsum[(size_t)t * EDIM + c], edge_attr[idx]);
  if (c == 0) atomicAdd(&deg[t], 1.0f);
}

__global__ void loop_attr_kernel(const float* __restrict__ lsum,
                                 const float* __restrict__ deg,
                                 float* __restrict__ lattr) {
  long long idx = (long long)blockIdx.x * blockDim.x + threadIdx.x;
  if (idx >= (long long)N_NODES * EDIM) return;
  int i = (int)(idx >> 4);
  lattr[idx] = lsum[idx] / fmaxf(deg[i], 1.0f);
}

// ---------------- fp32 WMMA GEMM: C[M,N] = A[M,K] @ B[K,N] ----------------
// 4 wave32 per block; each wave owns a 16x16 C tile of a 16x64 strip.
// A 16x4 layout: lane L(<16) holds {A[M=L,K=0],A[M=L,K=1]}; lane L+16 holds K=2,3.
// B 4x16 layout: mirrored with N across lanes. C/D: vgpr r -> M=r+(lane>=16?8:0), N=lane&15.
__global__ __launch_bounds__(128) void gemm_wmma_f32(
    const float* __restrict__ A, const float* __restrict__ B, float* __restrict__ C,
    int M, int N, int K, int transB) {
  __shared__ float As[16][68];
  __shared__ float Bs[64][72];
  const int tid  = threadIdx.x;
  const int lane = tid & 31;
  const int wave = tid >> 5;
  const int hi   = lane >> 4;      // 0 or 1 (half-wave)
  const int l16  = lane & 15;
  const int m0   = blockIdx.y * 16;
  const int n0   = blockIdx.x * 64;

  v8f acc = {0.f, 0.f, 0.f, 0.f, 0.f, 0.f, 0.f, 0.f};

  for (int kb = 0; kb < K; kb += 64) {
    for (int i = tid; i < 16 * 64; i += 128) {
      int r = i >> 6, c = i & 63;
      int gr = m0 + r, gc = kb + c;
      As[r][c] = (gr < M && gc < K) ? A[(size_t)gr * K + gc] : 0.0f;
    }
    for (int i = tid; i < 64 * 64; i += 128) {
      int r = i >> 6, c = i & 63;
      int gk = kb + r, gn = n0 + c;
      float v = 0.0f;
      if (gk < K && gn < N)
        v = transB ? B[(size_t)gn * K + gk] : B[(size_t)gk * N + gn];
      Bs[r][c] = v;
    }
    __syncthreads();
    const int bn = wave * 16 + l16;
    #pragma unroll
    for (int k4 = 0; k4 < 64; k4 += 4) {
      v2f a, b;
      a[0] = As[l16][k4 + hi * 2 + 0];
      a[1] = As[l16][k4 + hi * 2 + 1];
      b[0] = Bs[k4 + hi * 2 + 0][bn];
      b[1] = Bs[k4 + hi * 2 + 1][bn];
      acc = __builtin_amdgcn_wmma_f32_16x16x4_f32(
          /*neg_a=*/false, a, /*neg_b=*/false, b,
          /*c_mod=*/(short)0, acc, /*reuse_a=*/false, /*reuse_b=*/false);
    }
    __syncthreads();
  }
  const int col = n0 + wave * 16 + l16;
  #pragma unroll
  for (int r = 0; r < 8; ++r) {
    int row = m0 + r + hi * 8;
    if (row < M && col < N) C[(size_t)row * N + col] = acc[r];
  }
}

// ---------------- GATv2 edge score + segment max ----------------
// wave per edge (8 edges per wave serial). ee = edge_attr @ We recomputed from LDS.
__global__ __launch_bounds__(128) void gatv2_score(
    const float* __restrict__ xl, const float* __restrict__ xr,
    const int* __restrict__ src, const int* __restrict__ dst,
    const float* __restrict__ edge_attr, const float* __restrict__ loop_attr,
    const float* __restrict__ We, const float* __restrict__ att,
    float* __restrict__ score, float* __restrict__ smax,
    int D, int HD, int edgesPerWave) {
  extern __shared__ float smem[];  // [16*HD] We then [HD] att
  const int tid  = threadIdx.x;
  const int lane = tid & 31;
  const int wave = tid >> 5;
  const int tot  = 16 * HD + HD;
  for (int i = tid; i < tot; i += 128)
    smem[i] = (i < 16 * HD) ? We[i] : att[i - 16 * HD];
  __syncthreads();
  const float* attS = smem + 16 * HD;

  long long ebase = ((long long)blockIdx.x * 4 + wave) * edgesPerWave;
  for (int ei = 0; ei < edgesPerWave; ++ei) {
    long long e = ebase + ei;
    if (e >= (long long)EN) return;
    int s, t; const float* ea;
    if (e < (long long)N_EDGES) { s = src[e]; t = dst[e]; ea = edge_attr + e * EDIM; }
    else { s = t = (int)(e - N_EDGES); ea = loop_attr + (size_t)s * EDIM; }
    float eav[EDIM];
    #pragma unroll
    for (int c = 0; c < EDIM; ++c) eav[c] = ea[c];
    const float* xls = xl + (size_t)s * HD;
    const float* xrt = xr + (size_t)t * HD;
    #pragma unroll
    for (int h = 0; h < H_HEADS; ++h) {
      float acc = 0.0f;
      for (int d = lane; d < D; d += 32) {
        int j = h * D + d;
        float ee = 0.0f;
        #pragma unroll
        for (int c = 0; c < EDIM; ++c) ee += eav[c] * smem[c * HD + j];
        float m = xls[j] + xrt[j] + ee;
        m = (m > 0.0f) ? m : 0.2f * m;          // leaky_relu(0.2)
        acc += attS[j] * m;
      }
      #pragma unroll
      for (int off = 16; off >= 1; off >>= 1) acc += __shfl_xor(acc, off, 32);
      if (lane == 0) {
        score[e * H_HEADS + h] = acc;
        atomicMaxF(&smax[(size_t)t * H_HEADS + h], acc);
      }
    }
  }
}

// ---------------- softmax numerator + segment denominator ----------------
__global__ void gatv2_alpha(const float* __restrict__ score, const int* __restrict__ dst,
                            const float* __restrict__ smax,
                            float* __restrict__ alpha, float* __restrict__ denom) {
  long long idx = (long long)blockIdx.x * blockDim.x + threadIdx.x;
  if (idx >= (long long)EN * H_HEADS) return;
  long long e = idx >> 3;
  int h = (int)(idx & 7);
  int t = (e < (long long)N_EDGES) ? dst[e] : (int)(e - N_EDGES);
  float a = __expf(score[idx] - smax[(size_t)t * H_HEADS + h]);
  alpha[idx] = a;
  atomicAdd(&denom[(size_t)t * H_HEADS + h], a);
}

// ---------------- weighted aggregation: out[dst] += alpha * xl[src] ----------------
__global__ __launch_bounds__(128) void gatv2_aggregate(
    const float* __restrict__ xl, const int* __restrict__ src, const int* __restrict__ dst,
    const float* __restrict__ alpha, const float* __restrict__ denom,
    float* __restrict__ out, int HD, int logD) {
  long long e = (long long)blockIdx.x * 4 + (threadIdx.x >> 5);
  if (e >= (long long)EN) return;
  int lane = threadIdx.x & 31;
  int s, t;
  if (e < (long long)N_EDGES) { s = src[e]; t = dst[e]; }
  else { s = t = (int)(e - N_EDGES); }
  const float* xls = xl + (size_t)s * HD;
  float* outt = out + (size_t)t * HD;
  for (int j = lane; j < HD; j += 32) {
    int h = j >> logD;
    float w = alpha[e * H_HEADS + h] / denom[(size_t)t * H_HEADS + h];
    atomicAdd(&outt[j], w * xls[j]);   // non-returning -> global_atomic_add_f32
  }
}

// layer1 epilogue: in-place h1 = tanh(agg + b1[j])  (concat=True, HD1=512)
__global__ void epi1_kernel(float* __restrict__ h1, const float* __restrict__ b1) {
  long long idx = (long long)blockIdx.x * blockDim.x + threadIdx.x;
  if (idx >= (long long)N_NODES * HD1) return;
  h1[idx] = tanhf(h1[idx] + b1[idx & (HD1 - 1)]);
}

// layer2 epilogue: h2 = tanh(mean_h(agg2) + b2)  (concat=False)
__global__ void epi2_kernel(const float* __restrict__ agg2, const float* __restrict__ b2,
                            float* __restrict__ h2) {
  long long idx = (long long)blockIdx.x * blockDim.x + threadIdx.x;
  if (idx >= (long long)N_NODES * D2) return;
  int i = (int)(idx >> 7);
  int d = (int)(idx & 127);
  float s = 0.0f;
  #pragma unroll
  for (int h = 0; h < H_HEADS; ++h) s += agg2[(size_t)i * HD2 + h * D2 + d];
  h2[idx] = tanhf(s * 0.125f + b2[d]);
}

// fused LSTM (single step, h0=c0=0) + FC; one wave32 per node, lane = hidden unit
__global__ __launch_bounds__(256) void lstm_fc_kernel(
    const float* __restrict__ gates, const float* __restrict__ b_ih,
    const float* __restrict__ b_hh, const float* __restrict__ fcW,
    const float* __restrict__ fcb, float* __restrict__ out) {
  int n = blockIdx.x * 8 + (threadIdx.x >> 5);
  int lane = threadIdx.x & 31;
  if (n >= N_NODES) return;
  const float* g = gates + (size_t)n * (4 * HID);
  float iv = g[lane]      + b_ih[lane]      + b_hh[lane];
  float gv = g[64 + lane] + b_ih[64 + lane] + b_hh[64 + lane];
  float ov = g[96 + lane] + b_ih[96 + lane] + b_hh[96 + lane];
  float c   = sigmoidf_(iv) * tanhf(gv);
  float hid = sigmoidf_(ov) * tanhf(c);
  float acc = hid * fcW[lane];
  #pragma unroll
  for (int off = 16; off >= 1; off >>= 1) acc += __shfl_xor(acc, off, 32);
  if (lane == 0) out[n] = acc + fcb[0];
}

// ---------------- host-side orchestration ----------------
extern "C" void kernel_launch(void* const* d_in, const int* in_sizes, int n_in,
                              void* d_out, int out_size, void* d_ws, size_t ws_size,
                              hipStream_t stream) {
  const float* x    = (const float*)d_in[0];
  const int*   eidx = (const int*)d_in[1];
  const float* eatt = (const float*)d_in[2];
  const float* Wl1  = (const float*)d_in[3];
  const float* Wr1  = (const float*)d_in[4];
  const float* We1  = (const float*)d_in[5];
  const float* att1 = (const float*)d_in[6];
  const float* b1   = (const float*)d_in[7];
  const float* Wl2  = (const float*)d_in[8];
  const float* Wr2  = (const float*)d_in[9];
  const float* We2  = (const float*)d_in[10];
  const float* att2 = (const float*)d_in[11];
  const float* b2   = (const float*)d_in[12];
  const float* W_ih = (const float*)d_in[13];
  // d_in[14] = W_hh (unused: h0 = 0)
  const float* b_ih = (const float*)d_in[15];
  const float* b_hh = (const float*)d_in[16];
  const float* fcW  = (const float*)d_in[17];
  const float* fcb  = (const float*)d_in[18];
  const int* src = eidx;
  const int* dst = eidx + N_EDGES;
  float* out = (float*)d_out;

  // workspace layout (floats). xl2 aliases xl1+xr1 (both dead by then).
  float* ws = (float*)d_ws;
  const size_t off_xl1   = 0;                                   // N*HD1
  const size_t off_xr1   = (size_t)N_NODES * HD1;               // N*HD1
  const size_t off_xl2   = 0;                                   // N*HD2 (alias)
  const size_t off_h1    = (size_t)N_NODES * HD2;               // N*HD1 (agg1 -> h1 in place)
  const size_t off_xr2   = off_h1 + (size_t)N_NODES * HD1;      // N*HD2
  const size_t off_agg2  = off_xr2 + (size_t)N_NODES * HD2;     // N*HD2
  const size_t off_h2    = off_agg2 + (size_t)N_NODES * HD2;    // N*D2
  const size_t off_gates = off_h2 + (size_t)N_NODES * D2;       // N*4*HID
  const size_t off_score = off_gates + (size_t)N_NODES * 4 * HID; // EN*8
  const size_t off_alpha = off_score + (size_t)EN * H_HEADS;    // EN*8
  const size_t off_smax  = off_alpha + (size_t)EN * H_HEADS;    // N*8
  const size_t off_denom = off_smax + (size_t)N_NODES * H_HEADS;// N*8
  const size_t off_lsum  = off_denom + (size_t)N_NODES * H_HEADS;// N*16
  const size_t off_lattr = off_lsum + (size_t)N_NODES * EDIM;   // N*16
  const size_t off_deg   = off_lattr + (size_t)N_NODES * EDIM;  // N

  const int EPW = 8;  // edges per wave in score kernel
  const int scoreBlocks = (EN + 4 * EPW - 1) / (4 * EPW);
  const int alphaBlocks = (int)(((long long)EN * H_HEADS + 255) / 256);
  const int aggBlocks   = (EN + 3) / 4;

  // ---- init accumulators (every call: graph-replay safe) ----
  hipMemsetAsync(ws + off_deg,  0, sizeof(float) * N_NODES, stream);
  hipMemsetAsync(ws + off_lsum, 0, sizeof(float) * N_NODES * EDIM, stream);
  hipMemsetAsync(ws + off_h1,   0, sizeof(float) * (size_t)N_NODES * HD1, stream);
  hipMemsetAsync(ws + off_denom,0, sizeof(float) * N_NODES * H_HEADS, stream);
  fill_kernel<<<(N_NODES * H_HEADS + 255) / 256, 256, 0, stream>>>(
      ws + off_smax, -1e30f, (long long)N_NODES * H_HEADS);

  // ---- self-loop edge attr (mean of incoming) ----
  deg_loop_kernel<<<(int)(((long long)N_EDGES * EDIM + 255) / 256), 256, 0, stream>>>(
      dst, eatt, ws + off_deg, ws + off_lsum);
  loop_attr_kernel<<<(N_NODES * EDIM + 255) / 256, 256, 0, stream>>>(
      ws + off_lsum, ws + off_deg, ws + off_lattr);

  // ---- layer 1 GEMMs (WMMA) ----
  {
    dim3 grid(HD1 / 64, (N_NODES + 15) / 16);
    gemm_wmma_f32<<<grid, 128, 0, stream>>>(x, Wl1, ws + off_xl1, N_NODES, HD1, F_IN, 0);
    gemm_wmma_f32<<<grid, 128, 0, stream>>>(x, Wr1, ws + off_xr1, N_NODES, HD1, F_IN, 0);
  }

  // ---- layer 1 attention ----
  gatv2_score<<<scoreBlocks, 128, (16 * HD1 + HD1) * sizeof(float), stream>>>(
      ws + off_xl1, ws + off_xr1, src, dst, eatt, ws + off_lattr, We1, att1,
      ws + off_score, ws + off_smax, D1, HD1, EPW);
  gatv2_alpha<<<alphaBlocks, 256, 0, stream>>>(
      ws + off_score, dst, ws + off_smax, ws + off_alpha, ws + off_denom);
  gatv2_aggregate<<<aggBlocks, 128, 0, stream>>>(
      ws + off_xl1, src, dst, ws + off_alpha, ws + off_denom, ws + off_h1, HD1, 6);
  epi1_kernel<<<(int)(((long long)N_NODES * HD1 + 255) / 256), 256, 0, stream>>>(
      ws + off_h1, b1);

  // ---- layer 2 GEMMs (WMMA) ----
  {
    dim3 grid(HD2 / 64, (N_NODES + 15) / 16);
    gemm_wmma_f32<<<grid, 128, 0, stream>>>(ws + off_h1, Wl2, ws + off_xl2, N_NODES, HD2, HD1, 0);
    gemm_wmma_f32<<<grid, 128, 0, stream>>>(ws + off_h1, Wr2, ws + off_xr2, N_NODES, HD2, HD1, 0);
  }

  // ---- reset softmax accumulators for layer 2 ----
  hipMemsetAsync(ws + off_denom, 0, sizeof(float) * N_NODES * H_HEADS, stream);
  hipMemsetAsync(ws + off_agg2,  0, sizeof(float) * (size_t)N_NODES * HD2, stream);
  fill_kernel<<<(N_NODES * H_HEADS + 255) / 256, 256, 0, stream>>>(
      ws + off_smax, -1e30f, (long long)N_NODES * H_HEADS);

  // ---- layer 2 attention ----
  gatv2_score<<<scoreBlocks, 128, (16 * HD2 + HD2) * sizeof(float), stream>>>(
      ws + off_xl2, ws + off_xr2, src, dst, eatt, ws + off_lattr, We2, att2,
      ws + off_score, ws + off_smax, D2, HD2, EPW);
  gatv2_alpha<<<alphaBlocks, 256, 0, stream>>>(
      ws + off_score, dst, ws + off_smax, ws + off_alpha, ws + off_denom);
  gatv2_aggregate<<<aggBlocks, 128, 0, stream>>>(
      ws + off_xl2, src, dst, ws + off_alpha, ws + off_denom, ws + off_agg2, HD2, 7);
  epi2_kernel<<<(int)(((long long)N_NODES * D2 + 255) / 256), 256, 0, stream>>>(
      ws + off_agg2, b2, ws + off_h2);

  // ---- LSTM gate GEMM (WMMA, B transposed) + fused LSTM/FC ----
  {
    dim3 grid((4 * HID) / 64, (N_NODES + 15) / 16);
    gemm_wmma_f32<<<grid, 128, 0, stream>>>(ws + off_h2, W_ih, ws + off_gates,
                                            N_NODES, 4 * HID, D2, 1);
  }
  lstm_fc_kernel<<<(N_NODES + 7) / 8, 256, 0, stream>>>(
      ws + off_gates, b_ih, b_hh, fcW, fcb, out);
}